// SpottingLoss_16887811408707
// MI455X (gfx1250) — compile-verified
//
#include <hip/hip_runtime.h>

typedef float v2f __attribute__((ext_vector_type(2)));
typedef float v8f __attribute__((ext_vector_type(8)));

#define LAMBDA_COORD 5.0f
#define LAMBDA_NOOBJ 0.5f
#define NBOX 64
#define NF 19
#define FPAD 32
#define RS 44   // LDS row stride (floats): 2*RS % 64 = 24, 8*RS % 64 = 32

// One workgroup (64 threads = 2 wave32) per batch element.
__global__ __launch_bounds__(64) void spotting_loss_kernel(
    const float* __restrict__ y_true, const float* __restrict__ y_pred,
    float* __restrict__ partial)
{
    __shared__ float s_yt[NBOX * RS];
    __shared__ float s_yp[NBOX * RS];
    __shared__ int   s_bestcol[NBOX];
    __shared__ int   s_rowact[NBOX];
    __shared__ int   s_colact[NBOX];
    __shared__ int   s_c[NBOX];        // matched column per row (permutation)
    __shared__ float s_red[NBOX];
    __shared__ int   s_nact;

    const int b    = blockIdx.x;
    const int tid  = threadIdx.x;      // 0..63
    const int lane = tid & 31;
    const int wave = tid >> 5;

    // ---- Stage y_true / y_pred rows into LDS, zero-padded to 32 features ----
    {
        const float* yt = y_true + ((size_t)b * NBOX + tid) * NF;
        const float* yp = y_pred + ((size_t)b * NBOX + tid) * NF;
        for (int f = 0; f < FPAD; ++f) {
            s_yt[tid * RS + f] = (f < NF) ? yt[f] : 0.0f;
            s_yp[tid * RS + f] = (f < NF) ? yp[f] : 0.0f;
        }
        s_colact[tid] = 1;
        s_c[tid] = 0;                  // argmax of an all-zero Permut row is 0
    }
    __syncthreads();

    const float a_row = s_yt[tid * RS + 0];   // alpha_i (exactly 0.0 or 1.0)
    const float x_row = s_yt[tid * RS + 1];   // x_i
    const float p_col = s_yp[tid * RS + 1];   // p_j

    // ---- Two greedy mutual-best matching phases (positives, then negatives) ----
    for (int ph = 0; ph < 2; ++ph) {
        s_rowact[tid] = (ph == 0) ? ((a_row > 0.5f) ? 1 : 0)
                                  : ((a_row > 0.5f) ? 0 : 1);
        __syncthreads();
        for (int it = 0; it < NBOX; ++it) {
            if (tid == 0) s_nact = 0;
            __syncthreads();
            if (s_rowact[tid]) atomicAdd(&s_nact, 1);
            __syncthreads();
            if (s_nact == 0) break;    // uniform; remaining steps are no-ops

            // Row argmax over active columns (thread == row). D1 strictly > 0,
            // so first-max over active entries == jnp.argmax over masked D2.
            if (s_rowact[tid]) {
                float best = -1.0f; int bj = 0;
                for (int j = 0; j < NBOX; ++j) {
                    if (s_colact[j]) {
                        float v = 1.0f - fabsf(x_row - s_yp[j * RS + 1]); // broadcast read
                        if (v > best) { best = v; bj = j; }
                    }
                }
                s_bestcol[tid] = bj;
            }
            __syncthreads();

            // Column pass (thread == column): best candidate row, mutual-best match.
            int bi = -1;
            if (s_colact[tid]) {
                float best = -1.0f;
                for (int i = 0; i < NBOX; ++i) {
                    if (s_rowact[i] && s_bestcol[i] == tid) {
                        float v = 1.0f - fabsf(s_yt[i * RS + 1] - p_col); // broadcast read
                        if (v > best) { best = v; bi = i; }
                    }
                }
            }
            __syncthreads();           // reads of rowact/bestcol done before commits
            if (bi >= 0) {             // each row is candidate of exactly one column
                s_c[bi] = tid;
                s_rowact[bi] = 0;
                s_colact[tid] = 0;
            }
            __syncthreads();
        }
    }

    // ---- Apply permutation with WMMA: Yperm = P @ Ypred, P[i][j] = (c[i]==j).
    // Bit-exact vs gather: every product is 1.0f * y or 0.  128 wmma / batch.
    float lsum = 0.0f;
    const int mrow  = lane & 15;       // 0..15
    const int khalf = lane >> 4;       // lanes 16..31 carry K+2,K+3 (A) / rows K+2,K+3 (B)
    for (int mi = 0; mi < 2; ++mi) {
        const int mbase = (wave * 2 + mi) * 16;
        const int ci = s_c[mbase + mrow];
        for (int nt = 0; nt < 2; ++nt) {
            const int nbase = nt * 16;
            v8f acc = {};
            for (int kt = 0; kt < 16; ++kt) {
                const int kb = kt * 4 + khalf * 2;
                v2f av, bv;
                av.x = (ci == kb    ) ? 1.0f : 0.0f;   // A: 16x4 f32, lanes hold K pairs
                av.y = (ci == kb + 1) ? 1.0f : 0.0f;
                bv.x = s_yp[(kb    ) * RS + nbase + mrow]; // B: 4x16, row-striped in lanes
                bv.y = s_yp[(kb + 1) * RS + nbase + mrow];
                acc = __builtin_amdgcn_wmma_f32_16x16x4_f32(
                          false, av, false, bv, (short)0, acc, false, false);
            }
            // D tile layout: VGPR r -> row mbase+r (lanes 0-15) / mbase+8+r (16-31)
            for (int r = 0; r < 8; ++r) {
                const int   i  = mbase + r + khalf * 8;
                const int   nf = nbase + mrow;
                const float y  = acc[r];
                const float ai = s_yt[i * RS + 0];
                const float t  = s_yt[i * RS + nf];   // zero-padded for nf >= 19
                float term;
                if (nf == 0)      term = (ai + (1.0f - ai) * LAMBDA_NOOBJ) * (ai - y) * (ai - y);
                else if (nf == 1) term = ai * LAMBDA_COORD * (t - y) * (t - y);
                else              term = ai * (t - y) * (t - y);
                lsum += term;
            }
        }
    }

    // Deterministic fixed-order block reduction -> per-batch partial.
    s_red[tid] = lsum;
    __syncthreads();
    if (tid == 0) {
        float s = 0.0f;
        for (int i = 0; i < NBOX; ++i) s += s_red[i];
        partial[b] = s;
    }
}

__global__ __launch_bounds__(256) void spotting_reduce_kernel(
    const float* __restrict__ partial, float* __restrict__ out, int n)
{
    __shared__ float s[256];
    float v = 0.0f;
    for (int i = threadIdx.x; i < n; i += 256) v += partial[i];  // fixed order
    s[threadIdx.x] = v;
    __syncthreads();
    if (threadIdx.x == 0) {
        float t = 0.0f;
        for (int i = 0; i < 256; ++i) t += s[i];
        out[0] = t;
    }
}

extern "C" void kernel_launch(void* const* d_in, const int* in_sizes, int n_in,
                              void* d_out, int out_size, void* d_ws, size_t ws_size,
                              hipStream_t stream) {
    (void)n_in; (void)out_size; (void)ws_size;
    const float* y_true = (const float*)d_in[0];
    const float* y_pred = (const float*)d_in[1];
    float* out     = (float*)d_out;
    float* partial = (float*)d_ws;                 // Bn floats of scratch
    const int Bn = in_sizes[0] / (NBOX * NF);      // 2048

    spotting_loss_kernel<<<Bn, 64, 0, stream>>>(y_true, y_pred, partial);
    spotting_reduce_kernel<<<1, 256, 0, stream>>>(partial, out, Bn);
}